// TemporalAttentionLayer_72138270703924
// MI455X (gfx1250) — compile-verified
//
#include <hip/hip_runtime.h>
#include <math.h>

constexpr int TT = 64;     // sequence length
constexpr int DD = 512;    // model dim
constexpr int NH = 16;     // heads
constexpr int G  = 4;      // batches per block (amortizes weight scans 4x)

constexpr int HSTR = 516;            // sH row stride (4 mod 64 banks)
constexpr int RSTR = 516;            // sR row stride (16 rows, 4 valid)
constexpr int QSTR = 36;             // per-head q slice stride (32 + 4 pad)
constexpr int QG   = NH * QSTR;      // 576 floats per g
constexpr int UG   = DD * NH + 16;   // 8208: per-g U/C stride (16 mod 64 banks)

constexpr int OFF_H = 0;
constexpr int OFF_R = OFF_H + TT * HSTR;     // 33024
constexpr int OFF_Q = OFF_R + 16 * RSTR;     // 41280
constexpr int OFF_U = OFF_Q + G * QG;        // 43584
constexpr int OFF_S = OFF_U + G * UG;        // 76416
constexpr int LDS_FLOATS = OFF_S + TT * NH;  // 77440 floats = 309,760 B

typedef __attribute__((ext_vector_type(2))) float v2f;
typedef __attribute__((ext_vector_type(4))) float f4;
typedef __attribute__((ext_vector_type(8))) float v8f;

static __device__ __forceinline__ v8f wmma4(v2f a, v2f b, v8f c) {
    return __builtin_amdgcn_wmma_f32_16x16x4_f32(
        false, a, false, b, (short)0, c, false, false);
}

__global__ __launch_bounds__(256, 1)
void temporal_attn_lastrow_g4(const float* __restrict__ inputs,
                              const float* __restrict__ pos,
                              const float* __restrict__ Wq,
                              const float* __restrict__ Wk,
                              const float* __restrict__ Wv,
                              float* __restrict__ out)
{
    extern __shared__ float lds[];
    float* sH = lds + OFF_H;   // [64][516]   current batch's H = inputs+pos
    float* sR = lds + OFF_R;   // [16][516]   last-row inputs+pos (rows 0..3)
    float* sQ = lds + OFF_Q;   // [4][16][36] q per batch, per-head padded
    float* sU = lds + OFF_U;   // [4][512*16+16] U_g, overwritten by C_g
    float* sS = lds + OFF_S;   // [64][16]    scores -> softmax probs

    const int tid  = threadIdx.x;
    const int lane = tid & 31;
    const int wave = tid >> 5;
    const int m    = lane & 15;   // tile row / N index
    const int hi   = lane >> 4;   // half-wave K selector
    const long b0  = (long)blockIdx.x * G;

    // ---- S0: sR rows 0..3 = inputs[b,63,:] + pos[63,:]; rows 4..15 = 0 ------
    for (int i = tid; i < 16 * RSTR; i += 256) sR[i] = 0.f;
    __syncthreads();
    for (int i = tid; i < G * DD; i += 256) {
        int g = i >> 9, d = i & 511;
        sR[g * RSTR + d] =
            inputs[(b0 + g) * (size_t)(TT * DD) + 63 * DD + d] + pos[63 * DD + d];
    }
    __syncthreads();

    // ---- S1: Q = R @ Wq  (WMMA; M=16 rows, rows 0..3 = batches) -------------
    // 32 N-tiles over the 512 output columns, 4 tiles per wave; K=512.
    for (int t = 0; t < 4; ++t) {
        const int n0 = (wave * 4 + t) * 16;
        v8f acc = {};
        for (int k = 0; k < DD; k += 4) {
            const int kk = k + 2 * hi;
            v2f a, b;
            a[0] = sR[m * RSTR + kk];                       // A[M=m][K]
            a[1] = sR[m * RSTR + kk + 1];
            b[0] = Wq[(size_t)kk * DD + n0 + m];            // B[K][N=m] coalesced
            b[1] = Wq[(size_t)(kk + 1) * DD + n0 + m];
            acc = wmma4(a, b, acc);
        }
        if (hi == 0) {                                      // rows 0..3 live here
            #pragma unroll
            for (int r = 0; r < G; ++r) {
                const int e = n0 + m;
                sQ[r * QG + (e >> 5) * QSTR + (e & 31)] = acc[r];
            }
        }
    }
    __syncthreads();

    // ---- S2: U_g[d][h] = Wk[d, h-slice] . q_g[h-slice]  (per-head WMMA) -----
    // Jobs = 32 d-tiles x 16 heads; M=16 d-rows, N=g (4 valid), K=32.
    {
        const int   gl    = m & 3;
        const float vmask = (m < G) ? 1.f : 0.f;
        for (int t = 0; t < 64; ++t) {
            const int job = wave * 64 + t;
            const int dt  = job >> 4;        // h fastest: Wk lines reused in L0
            const int h   = job & 15;
            const int d0  = dt * 16;
            v8f acc = {};
            #pragma unroll
            for (int k = 0; k < 32; k += 4) {
                const int kk = k + 2 * hi;
                v2f a, b;
                a[0] = Wk[(size_t)(d0 + m) * DD + 32 * h + kk];       // A[M=d][K=j]
                a[1] = Wk[(size_t)(d0 + m) * DD + 32 * h + kk + 1];
                b[0] = vmask * sQ[gl * QG + h * QSTR + kk];           // B[K=j][N=g]
                b[1] = vmask * sQ[gl * QG + h * QSTR + kk + 1];
                acc = wmma4(a, b, acc);
            }
            if (m < G) {
                #pragma unroll
                for (int r = 0; r < 8; ++r)
                    sU[m * UG + (d0 + r + 8 * hi) * NH + h] = acc[r];
            }
        }
    }
    __syncthreads();

    // ---- S3: per batch g: load H, S = H@U_g, softmax, C_g = H^T@P -----------
    const float scale = 0.044194173824159216f;   // 1/sqrt(512)
    for (int g = 0; g < G; ++g) {
        // H_g = inputs[b0+g] + pos  (stream 128 KB into LDS)
        {
            const f4* in4 = (const f4*)(inputs + (b0 + g) * (size_t)(TT * DD));
            const f4* pe4 = (const f4*)pos;
            for (int i = tid; i < TT * DD / 4; i += 256) {
                f4 x = in4[i], p = pe4[i];
                int t = i >> 7, d = (i & 127) << 2;
                float* dst = sH + t * HSTR + d;
                dst[0] = x.x + p.x; dst[1] = x.y + p.y;
                dst[2] = x.z + p.z; dst[3] = x.w + p.w;
            }
            if (g + 1 < G) {   // pull next batch's tile toward L2 while we compute
                const float* nxt = inputs + (b0 + g + 1) * (size_t)(TT * DD);
                #pragma unroll
                for (int it = 0; it < 4; ++it)
                    __builtin_prefetch(nxt + (size_t)(tid + it * 256) * 32, 0, 1);
            }
        }
        __syncthreads();

        // S = (H @ U_g) * scale  [64x16]; waves 0..3, one 16-row tile each
        if (wave < 4) {
            const int t0 = wave * 16;
            v8f acc = {};
            for (int k = 0; k < DD; k += 4) {
                const int kk = k + 2 * hi;
                v2f a, b;
                a[0] = sH[(t0 + m) * HSTR + kk];
                a[1] = sH[(t0 + m) * HSTR + kk + 1];
                b[0] = sU[g * UG + kk * NH + m];
                b[1] = sU[g * UG + (kk + 1) * NH + m];
                acc = wmma4(a, b, acc);
            }
            #pragma unroll
            for (int r = 0; r < 8; ++r)
                sS[(t0 + r + 8 * hi) * NH + m] = acc[r] * scale;
        }
        __syncthreads();

        // softmax over t per head (last-row query: causal mask is all-ones)
        if (tid < NH) {
            float mx = -INFINITY;
            for (int t = 0; t < TT; ++t) mx = fmaxf(mx, sS[t * NH + tid]);
            float sum = 0.f;
            for (int t = 0; t < TT; ++t) {
                float e = expf(sS[t * NH + tid] - mx);
                sS[t * NH + tid] = e;
                sum += e;
            }
            float inv = 1.f / sum;
            for (int t = 0; t < TT; ++t) sS[t * NH + tid] *= inv;
        }
        __syncthreads();

        // C_g = H^T @ P  [512x16]; 32 d-tiles, 4 per wave; overwrites U_g
        for (int t = 0; t < 4; ++t) {
            const int d0 = (wave * 4 + t) * 16;
            v8f acc = {};
            for (int k = 0; k < TT; k += 4) {
                const int kk = k + 2 * hi;
                v2f a, b;
                a[0] = sH[kk * HSTR + d0 + m];        // A[M=d][K=t] = H[t][d]
                a[1] = sH[(kk + 1) * HSTR + d0 + m];
                b[0] = sS[kk * NH + m];               // B[K=t][N=h]
                b[1] = sS[(kk + 1) * NH + m];
                acc = wmma4(a, b, acc);
            }
            #pragma unroll
            for (int r = 0; r < 8; ++r)
                sU[g * UG + (d0 + r + 8 * hi) * NH + m] = acc[r];
        }
        __syncthreads();
    }

    // ---- S4: out[g][e] = sum_d C_g[d][h(e)] * Wv[d][e]  (per-head WMMA) -----
    // Jobs = 16 heads x 2 col-tiles; M=g rows (4 valid), K=512, N=16 cols.
    {
        const int   gl    = m & 3;
        const float vmask = (m < G) ? 1.f : 0.f;
        for (int t = 0; t < 4; ++t) {
            const int job = wave * 4 + t;
            const int h   = job >> 1;
            const int n0  = (job & 1) * 16;
            v8f acc = {};
            for (int k = 0; k < DD; k += 4) {
                const int kk = k + 2 * hi;
                v2f a, b;
                a[0] = vmask * sU[gl * UG + kk * NH + h];        // A[M=g][K=d]
                a[1] = vmask * sU[gl * UG + (kk + 1) * NH + h];
                b[0] = Wv[(size_t)kk * DD + 32 * h + n0 + m];    // B[K=d][N] coalesced
                b[1] = Wv[(size_t)(kk + 1) * DD + 32 * h + n0 + m];
                acc = wmma4(a, b, acc);
            }
            if (hi == 0) {
                #pragma unroll
                for (int r = 0; r < G; ++r)
                    out[(b0 + r) * (size_t)DD + 32 * h + n0 + m] = acc[r];
            }
        }
    }
}

extern "C" void kernel_launch(void* const* d_in, const int* in_sizes, int n_in,
                              void* d_out, int out_size, void* d_ws, size_t ws_size,
                              hipStream_t stream) {
    const float* inputs = (const float*)d_in[0];  // [B,64,512]
    const float* pos    = (const float*)d_in[1];  // [64,512]
    const float* Wq     = (const float*)d_in[2];  // [512,512]
    const float* Wk     = (const float*)d_in[3];
    const float* Wv     = (const float*)d_in[4];
    float* out          = (float*)d_out;          // [B,512]
    (void)d_ws; (void)ws_size; (void)n_in; (void)out_size;

    const int B = in_sizes[0] / (TT * DD);        // 2048
    const size_t smem = (size_t)LDS_FLOATS * sizeof(float);  // ~302 KB

    hipFuncSetAttribute((const void*)temporal_attn_lastrow_g4,
                        hipFuncAttributeMaxDynamicSharedMemorySize, (int)smem);

    temporal_attn_lastrow_g4<<<B / G, 256, smem, stream>>>(
        inputs, pos, Wq, Wk, Wv, out);
}